// Vec2Im_4982162063687
// MI455X (gfx1250) — compile-verified
//
#include <hip/hip_runtime.h>
#include <hip/hip_bf16.h>

typedef __attribute__((ext_vector_type(2))) float v2f;
typedef __attribute__((ext_vector_type(4))) float v4f;
typedef __attribute__((ext_vector_type(8))) float v8f;

#define B_DIM   1024
#define N_DIM   34
#define P_TOTAL (B_DIM * N_DIM)   // 34816
#define H_IMG   256
#define W_IMG   256
#define MAX_RX  34
#define NCAT    5
#define GG      23                // number of basis functions (NUM + K)
#define KPAD    24                // padded K for 6 chunks of 4
#define KORD    8
#define NGRID   32                // NUM + 2K + 1 grid points

#define TPB     128               // 4 waves of 32
#define WAVES   (TPB / 32)
#define NBLK    (P_TOTAL / TPB)   // 272, exact

// ---------------------------------------------------------------------------
// Uniform B-spline grid: grid[i] = RLO + (i - K) * h, h = (RHI-RLO)/NUM
// ---------------------------------------------------------------------------
__device__ __forceinline__ float grid_at(int i) {
    const float h = 4.0f / 15.0f;
    return -2.0f + (float)(i - KORD) * h;
}

// Cox-de-Boor recursion, degree 8, 31 intervals -> 23 basis values (+1 pad 0)
__device__ __forceinline__ void bspline_basis23(float x, float* out24) {
    const float h = 4.0f / 15.0f;
    float b[NGRID - 1];
#pragma unroll
    for (int i = 0; i < NGRID - 1; ++i)
        b[i] = (x >= grid_at(i) && x < grid_at(i + 1)) ? 1.0f : 0.0f;
#pragma unroll
    for (int d = 1; d <= KORD; ++d) {
        float inv = 1.0f / ((float)d * h);
#pragma unroll
        for (int i = 0; i < NGRID - 1 - d; ++i) {
            float left  = (x - grid_at(i)) * inv;
            float right = (grid_at(i + d + 1) - x) * inv;
            b[i] = left * b[i] + right * b[i + 1];
        }
    }
#pragma unroll
    for (int k = 0; k < GG; ++k) out24[k] = b[k];
    out24[GG] = 0.0f;
}

__device__ __forceinline__ float silu(float x) {
    return x / (1.0f + expf(-x));
}

// ---------------------------------------------------------------------------
// Kernel 1: zero-fill 512MB of output with NT b128 streaming stores.
// This is the entire roofline cost of the problem (~23us at 23.3 TB/s).
// ---------------------------------------------------------------------------
__global__ void __launch_bounds__(256) fill_zero_kernel(v4f* __restrict__ out, int n4) {
    int stride = gridDim.x * blockDim.x;
    v4f z = {0.0f, 0.0f, 0.0f, 0.0f};
    for (int i = blockIdx.x * blockDim.x + threadIdx.x; i < n4; i += stride)
        __builtin_nontemporal_store(z, out + i);
}

// ---------------------------------------------------------------------------
// Kernel 2: two-stage KAN spline per point.
// Stage 1 (device spline) is computed as a dense GEMM vs ALL 34 device rows
// via V_WMMA_F32_16X16X4_F32 (f32 precision), then row dev_id is selected.
// Stage 2 (cat spline, only 5 rows) stays scalar.
// p2 results go to workspace for a deterministic scatter pass.
// ---------------------------------------------------------------------------
__global__ void __launch_bounds__(TPB) spline_kernel(
    const float* __restrict__ x_vecs,
    const float* __restrict__ dcoef,   // (34, 23)
    const float* __restrict__ dbase,   // (34,)
    const float* __restrict__ ccoef,   // (5, 23)
    const float* __restrict__ cbase,   // (5,)
    float* __restrict__ pws,           // (P_TOTAL,) stage-2 results
    float* __restrict__ out_direct,    // only used when mode==1
    int mode)
{
    __shared__ float sCoef[48 * KPAD];                 // 48 rows (34 + pad) x 24
    __shared__ float sBx[TPB * KPAD];                  // per-point basis row
    __shared__ float sD[WAVES * 2 * 48 * 16];          // GEMM results per wave/group

    const int t = threadIdx.x;

    // Cooperative load of device coef into LDS, zero-padded to 48x24
    for (int i = t; i < 48 * KPAD; i += TPB) {
        int r = i / KPAD, k = i % KPAD;
        sCoef[i] = (r < MAX_RX && k < GG) ? dcoef[r * GG + k] : 0.0f;
    }

    const int g = blockIdx.x * TPB + t;                // point index (exact cover)
    const float x   = x_vecs[g * 5 + 0];
    const int devId = (int)x_vecs[g * 5 + 3];
    const int catId = (int)x_vecs[g * 5 + 4];

    float bx[KPAD];
    bspline_basis23(x, bx);
#pragma unroll
    for (int k = 0; k < KPAD; ++k) sBx[t * KPAD + k] = bx[k];
    __syncthreads();

    const int wave = t >> 5;
    const int lane = t & 31;
    const int half = lane >> 4;       // 0: lanes 0-15, 1: lanes 16-31
    const int l16  = lane & 15;

    float val;
#if __has_builtin(__builtin_amdgcn_wmma_f32_16x16x4_f32)
    // D[r, n] = sum_k coef[r, k] * Bx[n, k]  for a 16-point group n, 16-row tile r.
    // A (16x4 f32): vgpr0 = (M=l16, K=2*half), vgpr1 = K=2*half+1  (per ISA layout)
    // B (4x16 f32): vgpr0 = (K=2*half, N=l16), vgpr1 = K=2*half+1
#pragma unroll
    for (int grp = 0; grp < 2; ++grp) {
        const int prow = wave * 32 + grp * 16 + l16;   // point row inside block
#pragma unroll
        for (int tt = 0; tt < 3; ++tt) {               // row tiles: 0-15,16-31,32-47
            v8f acc = {};
#pragma unroll
            for (int kc = 0; kc < 6; ++kc) {           // K = 24 in chunks of 4
                const int kb = kc * 4 + 2 * half;
                v2f a, bb;
                a[0]  = sCoef[(tt * 16 + l16) * KPAD + kb + 0];
                a[1]  = sCoef[(tt * 16 + l16) * KPAD + kb + 1];
                bb[0] = sBx[prow * KPAD + kb + 0];
                bb[1] = sBx[prow * KPAD + kb + 1];
                acc = __builtin_amdgcn_wmma_f32_16x16x4_f32(
                        false, a, false, bb, (short)0, acc, false, false);
            }
            // C/D layout: vgpr r holds (M = r + 8*half, N = l16)
#pragma unroll
            for (int r = 0; r < 8; ++r) {
                const int row = tt * 16 + r + 8 * half;
                sD[((wave * 2 + grp) * 48 + row) * 16 + l16] = acc[r];
            }
        }
    }
    __syncthreads();
    // This thread's point lives in group `half`, column `l16` of its wave.
    val = sD[((wave * 2 + half) * 48 + devId) * 16 + l16];
#else
    // Scalar fallback: 23-term dot against the selected row.
    val = 0.0f;
#pragma unroll
    for (int k = 0; k < GG; ++k) val += sCoef[devId * KPAD + k] * bx[k];
#endif

    const float p1 = dbase[devId] * silu(x) + val;

    // Stage 2: category spline (5 rows -> scalar dot)
    float bx2[KPAD];
    bspline_basis23(p1, bx2);
    float acc2 = 0.0f;
#pragma unroll
    for (int k = 0; k < GG; ++k) acc2 += ccoef[catId * GG + k] * bx2[k];
    const float p2 = cbase[catId] * silu(p1) + acc2;

    if (mode == 0) {
        pws[g] = p2;
    } else {
        // Fallback direct scatter (only if workspace is too small)
        const int b  = g / N_DIM;
        const int cx = __float2int_rn(x_vecs[g * 5 + 1]);
        const int cy = __float2int_rn(x_vecs[g * 5 + 2]);
        size_t o = (size_t)b * 2 * H_IMG * W_IMG + (size_t)cy * W_IMG + cx;
        out_direct[o] = p2;
        out_direct[o + H_IMG * W_IMG] = x;
    }
}

// ---------------------------------------------------------------------------
// Kernel 3: deterministic scatter. One thread per batch image walks its 34
// points in index order, so duplicate coordinates resolve last-index-wins
// identically on every replay.
// ---------------------------------------------------------------------------
__global__ void scatter_kernel(const float* __restrict__ x_vecs,
                               const float* __restrict__ pws,
                               float* __restrict__ out)
{
    int b = blockIdx.x * blockDim.x + threadIdx.x;
    if (b >= B_DIM) return;
    const size_t base0 = (size_t)b * 2 * H_IMG * W_IMG;
    for (int n = 0; n < N_DIM; ++n) {
        const int g  = b * N_DIM + n;
        const int cx = __float2int_rn(x_vecs[g * 5 + 1]);
        const int cy = __float2int_rn(x_vecs[g * 5 + 2]);
        const size_t o = base0 + (size_t)cy * W_IMG + cx;
        out[o] = pws[g];                                 // channel 0: p
        out[o + H_IMG * W_IMG] = x_vecs[g * 5 + 0];      // channel 1: power
    }
}

extern "C" void kernel_launch(void* const* d_in, const int* in_sizes, int n_in,
                              void* d_out, int out_size, void* d_ws, size_t ws_size,
                              hipStream_t stream) {
    const float* x_vecs = (const float*)d_in[0];
    const float* dcoef  = (const float*)d_in[1];
    const float* dbase  = (const float*)d_in[2];
    const float* ccoef  = (const float*)d_in[3];
    const float* cbase  = (const float*)d_in[4];
    float* out = (float*)d_out;

    // 1) Zero-fill 512MB output (the roofline cost) with NT b128 stores.
    const int n4 = out_size / 4;                 // 33,554,432 float4
    fill_zero_kernel<<<2048, 256, 0, stream>>>((v4f*)out, n4);

    // 2) Spline compute; 3) deterministic scatter.
    const bool haveWs = (ws_size >= (size_t)P_TOTAL * sizeof(float));
    float* pws = (float*)d_ws;
    spline_kernel<<<NBLK, TPB, 0, stream>>>(x_vecs, dcoef, dbase, ccoef, cbase,
                                            pws, out, haveWs ? 0 : 1);
    if (haveWs) {
        scatter_kernel<<<(B_DIM + 255) / 256, 256, 0, stream>>>(x_vecs, pws, out);
    }
}